// RelativeAttention_45200235823811
// MI455X (gfx1250) — compile-verified
//
#include <hip/hip_runtime.h>

// Problem constants (from setup_inputs): B=4,H=8,S=1024,D=64,CUT=64
#define TQ    32            // query rows per workgroup
#define SEQ   1024
#define DIM   64
#define NHEAD 32            // B*H
#define RCUT  64
#define RTAB  129           // 2*RCUT+1
#define KB    128           // keys per staged block
#define NKB   (SEQ / KB)    // 8 blocks

// LDS layout (floats). Strides padded for bank-conflict-free WMMA operand loads:
//  66 % 64 = 2, 148 % 64 = 20, 1028 % 64 = 4, 76 % 64 = 12.
#define SQ_STRIDE   66
#define SRS_STRIDE  148
#define SRC_STRIDE  148
#define SS_STRIDE   1028
#define KV_STRIDE   76                            // multiple of 4 floats (b128 aligned)
#define KVBUF       (KB * KV_STRIDE)              // 9728 floats per buffer
#define OFF_SQ   0
#define OFF_SRS  (OFF_SQ  + TQ * SQ_STRIDE)       // 2112
#define OFF_SRC  (OFF_SRS + TQ * SRS_STRIDE)      // 6848
#define OFF_SS   (OFF_SRC + TQ * SRC_STRIDE)      // 11584
#define OFF_KV   (OFF_SS  + TQ * SS_STRIDE)       // 44480
#define SMEM_FLOATS (OFF_KV + 2 * KVBUF)          // 63936 floats
#define SMEM_BYTES  (SMEM_FLOATS * 4)             // 255744 B (< 320 KB WGP LDS)

typedef float v2f __attribute__((ext_vector_type(2)));
typedef float v4f __attribute__((ext_vector_type(4)));
typedef float v8f __attribute__((ext_vector_type(8)));
typedef int   v4i __attribute__((ext_vector_type(4)));

// ---- CDNA5 async global->LDS path (ASYNCcnt), with synchronous fallback ----
#if __has_builtin(__builtin_amdgcn_global_load_async_to_lds_b128) && \
    __has_builtin(__builtin_amdgcn_s_wait_asynccnt)
#define HAVE_ASYNC 1
#define WAIT_ASYNC(n) __builtin_amdgcn_s_wait_asynccnt(n)
#else
#define HAVE_ASYNC 0
#define WAIT_ASYNC(n)
#endif

__device__ __forceinline__ void cp_b128(float* dst_lds, const float* src_g) {
#if HAVE_ASYNC
  // Builtin signature (from hipcc diagnostic): (int4 AS1*, int4 AS3*, imm, imm)
  __builtin_amdgcn_global_load_async_to_lds_b128(
      (__attribute__((address_space(1))) v4i*)src_g,
      (__attribute__((address_space(3))) v4i*)dst_lds, 0, 0);
#else
  *(v4f*)dst_lds = *(const v4f*)src_g;
#endif
}

// Stage one [128 x 64] f32 block (K or V rows) into an LDS buffer with padded
// stride. 2048 x b128 transfers, 8 per thread, fully coalesced (16 lanes cover
// one 256B row).
__device__ __forceinline__ void stage_block(const float* __restrict__ srcbase,
                                            float* __restrict__ dstbuf, int tid) {
#pragma unroll
  for (int t = 0; t < 8; ++t) {
    int idx = tid + (t << 8);        // 0..2047
    int row = idx >> 4;
    int c4  = (idx & 15) << 2;
    cp_b128(dstbuf + row * KV_STRIDE + c4, srcbase + row * DIM + c4);
  }
}

// D = A(16x4 f32) * B(4x16 f32) + C(16x16 f32)  — CDNA5 V_WMMA_F32_16X16X4_F32
__device__ __forceinline__ v8f wmma4(v2f a, v2f b, v8f c) {
  return __builtin_amdgcn_wmma_f32_16x16x4_f32(false, a, false, b, (short)0, c,
                                               false, false);
}

extern "C" __global__ __launch_bounds__(256, 1)
void relattn_kernel(const float* __restrict__ Q, const float* __restrict__ K,
                    const float* __restrict__ V, const float* __restrict__ WK,
                    const float* __restrict__ WV, float* __restrict__ O)
{
  extern __shared__ float sm[];
  float* sQ  = sm + OFF_SQ;    // [TQ][66]    staged Q tile
  float* sRS = sm + OFF_SRS;   // [TQ][148]   rel_s[i][r] = q_i . w_k[r]
  float* sRC = sm + OFF_SRC;   // [TQ][148]   c[i][r] = sum_j p[i,j] [clip(j-i)=r-64]
  float* sS  = sm + OFF_SS;    // [TQ][1028]  scores -> probabilities
  float* sKV = sm + OFF_KV;    // [2][128][76] double-buffered K/V block

  const int tid  = threadIdx.x;
  const int wave = tid >> 5;
  const int lane = tid & 31;
  const int hrow = lane & 15;          // row (A) / col (B,C,D) index within tile
  const int kk   = (lane >> 4) << 1;   // K-dim sub-offset {0,2}
  const int vro  = (lane >> 4) << 3;   // C/D row offset {0,8}
  const int q0   = blockIdx.x * TQ;
  const int bh   = blockIdx.y;

  const float* Qp = Q + ((size_t)bh * SEQ + q0) * DIM;
  const float* Kp = K + (size_t)bh * SEQ * DIM;
  const float* Vp = V + (size_t)bh * SEQ * DIM;

  // ---------- Phase 0: stage Q tile into LDS; zero rel accumulator ----------
  for (int idx = tid; idx < TQ * DIM / 2; idx += 256) {
    int i = idx >> 5;               // row 0..31
    int d = (idx & 31) << 1;        // even col 0..62
    *(v2f*)&sQ[i * SQ_STRIDE + d] = *(const v2f*)&Qp[i * DIM + d];
  }
  for (int idx = tid; idx < TQ * SRC_STRIDE; idx += 256)
    sRC[idx] = 0.0f;
  // Prefetch K block 0 into buffer 0 while phase 1 runs.
  stage_block(Kp, sKV, tid);
  __syncthreads();

  // ---------- Phase 1: rel_s = Q_tile @ WK^T   (2 x 9 tiles of 16x16) ------
  for (int tile = wave; tile < 18; tile += 8) {
    const int m0 = (tile / 9) * 16;
    const int r0 = (tile % 9) * 16;
    const int rc = min(r0 + hrow, RTAB - 1);   // clamp pad cols (129..143)
    v8f acc = {};
#pragma unroll
    for (int k0 = 0; k0 < DIM; k0 += 4) {
      v2f a = *(const v2f*)&sQ[(m0 + hrow) * SQ_STRIDE + k0 + kk];
      v2f b = *(const v2f*)&WK[rc * DIM + k0 + kk];
      acc = wmma4(a, b, acc);
    }
#pragma unroll
    for (int v = 0; v < 8; ++v)
      sRS[(m0 + v + vro) * SRS_STRIDE + r0 + hrow] = acc[v];
  }
  __syncthreads();

  // ---------- Phase 2: scores = Q @ K^T + gathered relative bias ----------
  // Double-buffered async staging of 128-key K blocks; each block read once
  // from L2 instead of once per row-group wave.
  for (int kb = 0; kb < NKB; ++kb) {
    float* buf = sKV + (kb & 1) * KVBUF;
    if (kb + 1 < NKB) {
      stage_block(Kp + (size_t)(kb + 1) * KB * DIM, sKV + ((kb + 1) & 1) * KVBUF, tid);
      WAIT_ASYNC(8);    // only the 8 just-issued (next block) may be outstanding
    } else {
      WAIT_ASYNC(0);
    }
    __syncthreads();
    for (int t = wave; t < 16; t += 8) {       // 2 row-groups x 8 local key tiles
      const int m0  = (t >> 3) * 16;
      const int jl0 = (t & 7) * 16;
      v8f acc = {};
#pragma unroll
      for (int k0 = 0; k0 < DIM; k0 += 4) {
        v2f a = *(const v2f*)&sQ[(m0 + hrow) * SQ_STRIDE + k0 + kk];
        v2f b = *(const v2f*)&buf[(jl0 + hrow) * KV_STRIDE + k0 + kk];
        acc = wmma4(a, b, acc);
      }
      const int jg = kb * KB + jl0 + hrow;
#pragma unroll
      for (int v = 0; v < 8; ++v) {
        int il  = m0 + v + vro;
        int rel = jg - (q0 + il);
        rel = rel < -RCUT ? -RCUT : (rel > RCUT ? RCUT : rel);
        sS[il * SS_STRIDE + jg] = acc[v] + sRS[il * SRS_STRIDE + rel + RCUT];
      }
    }
    __syncthreads();
  }

  // Prefetch V block 0; async engine overlaps with the softmax phase.
  stage_block(Vp, sKV, tid);

  // ---------- Phase 3: row softmax + band-histogram of probabilities -------
  {
    const int row = tid >> 3;     // 8 threads per row, same wave
    const int sub = tid & 7;
    float* srow = &sS[row * SS_STRIDE];
    float mx = -3.402823466e38f;
    for (int j = sub; j < SEQ; j += 8) mx = fmaxf(mx, srow[j]);
    for (int m = 4; m; m >>= 1) mx = fmaxf(mx, __shfl_xor(mx, m, 8));
    float sum = 0.0f;
    for (int j = sub; j < SEQ; j += 8) {
      float e = __expf(srow[j] - mx);
      srow[j] = e;
      sum += e;
    }
    for (int m = 4; m; m >>= 1) sum += __shfl_xor(sum, m, 8);
    const float inv = 1.0f / sum;
    const int ig = q0 + row;
    float lo = 0.0f, hi = 0.0f;
    float* crow = &sRC[row * SRC_STRIDE];
    for (int j = sub; j < SEQ; j += 8) {
      float p = srow[j] * inv;
      srow[j] = p;                       // normalized probability back to LDS
      int rel = j - ig;
      if (rel <= -RCUT)      lo += p;    // clipped left edge  -> r = 0
      else if (rel >= RCUT)  hi += p;    // clipped right edge -> r = 128
      else                   crow[rel + RCUT] = p;  // unique r per j: plain store
    }
    atomicAdd(&crow[0], lo);             // ds_add_f32
    atomicAdd(&crow[2 * RCUT], hi);
  }
  __syncthreads();

  // ---------- Phase 4: out = P @ V + c @ WV  (one 16x16 tile per wave) -----
  {
    const int m0 = (wave >> 2) * 16;
    const int n0 = (wave & 3) * 16;
    v8f acc = {};
    for (int kb = 0; kb < NKB; ++kb) {
      float* buf = sKV + (kb & 1) * KVBUF;
      if (kb + 1 < NKB) {
        stage_block(Vp + (size_t)(kb + 1) * KB * DIM, sKV + ((kb + 1) & 1) * KVBUF, tid);
        WAIT_ASYNC(8);
      } else {
        WAIT_ASYNC(0);
      }
      __syncthreads();
      const int ks = kb * KB;
      for (int k0 = 0; k0 < KB; k0 += 4) {
        v2f a = *(const v2f*)&sS[(m0 + hrow) * SS_STRIDE + ks + k0 + kk];
        v2f b;
        b.x = buf[(k0 + kk)     * KV_STRIDE + n0 + hrow];
        b.y = buf[(k0 + kk + 1) * KV_STRIDE + n0 + hrow];
        acc = wmma4(a, b, acc);
      }
      __syncthreads();
    }
    // rel-value tail: out += c @ WV   (129 rounded to 132; tail A rows are 0)
#pragma unroll
    for (int k0 = 0; k0 < 132; k0 += 4) {
      v2f a = *(const v2f*)&sRC[(m0 + hrow) * SRC_STRIDE + k0 + kk];
      int r0c = min(k0 + kk,     RTAB - 1);
      int r1c = min(k0 + kk + 1, RTAB - 1);
      v2f b;
      b.x = WV[r0c * DIM + n0 + hrow];
      b.y = WV[r1c * DIM + n0 + hrow];
      acc = wmma4(a, b, acc);
    }
    float* Op = O + ((size_t)bh * SEQ + q0) * DIM;
#pragma unroll
    for (int v = 0; v < 8; ++v)
      Op[(m0 + v + vro) * DIM + n0 + hrow] = acc[v];
  }
}

extern "C" void kernel_launch(void* const* d_in, const int* in_sizes, int n_in,
                              void* d_out, int out_size, void* d_ws, size_t ws_size,
                              hipStream_t stream) {
  const float* Q  = (const float*)d_in[0];
  const float* K  = (const float*)d_in[1];
  const float* V  = (const float*)d_in[2];
  const float* WK = (const float*)d_in[3];
  const float* WV = (const float*)d_in[4];
  float* O = (float*)d_out;

  (void)in_sizes; (void)n_in; (void)out_size; (void)d_ws; (void)ws_size;

  (void)hipFuncSetAttribute((const void*)relattn_kernel,
                            hipFuncAttributeMaxDynamicSharedMemorySize, SMEM_BYTES);

  dim3 grid(SEQ / TQ, NHEAD);   // 32 q-tiles x 32 (b,h) heads
  relattn_kernel<<<grid, 256, SMEM_BYTES, stream>>>(Q, K, V, WK, WV, O);
}